// IMUFeatureExtractor_11991548690772
// MI455X (gfx1250) — compile-verified
//
#include <hip/hip_runtime.h>
#include <math.h>

// IMU feature extraction: (B,7,T) f32 -> (B,16,T) f32
// Channels out: [imu(7), linear_acc(3), lam(1), jerk(1), ang_vel(3), ang_dist(1)]
// Bandwidth-bound elementwise + time-stencil kernel; wave32, b128 vmem, NT stores.

typedef float v4f __attribute__((ext_vector_type(4)));

#define TOLF 1e-8f
#define DTF  0.005f     // float32(1/200)
#define BLOCK 256
#define VEC   4
#define CHUNK (BLOCK * VEC)

__device__ __forceinline__ bool finitef(float x) {
    return fabsf(x) < __builtin_inff();
}

// Recompute lam (|linear_acc|) for a single global time index (halo).
__device__ __forceinline__ float lam_at(const float* __restrict__ imu_b, int T, int t) {
    float ax = imu_b[0 * T + t];
    float ay = imu_b[1 * T + t];
    float az = imu_b[2 * T + t];
    float w  = imu_b[3 * T + t];
    float x  = imu_b[4 * T + t];
    float y  = imu_b[5 * T + t];
    float z  = imu_b[6 * T + t];
    float n   = sqrtf(x * x + y * y + z * z + w * w);
    bool  v   = n > TOLF;
    float inv = 1.0f / (v ? n : 1.0f);
    float xn = x * inv, yn = y * inv, zn = z * inv, wn = w * inv;
    float gx = v ? 19.62f * (xn * zn - wn * yn) : 0.0f;
    float gy = v ? 19.62f * (wn * xn + yn * zn) : 0.0f;
    float gz = v ? (9.81f - 19.62f * (xn * xn + yn * yn)) : 0.0f;
    float lx = ax - gx, ly = ay - gy, lz = az - gz;  // g==0 when !v => linear=acc
    return sqrtf(lx * lx + ly * ly + lz * lz);
}

__global__ __launch_bounds__(BLOCK) void imu_feat_kernel(
    const float* __restrict__ imu, float* __restrict__ out, int T) {

    __shared__ float s_lam[BLOCK + 1];                 // s_lam[i+1] = lam of thread i's last elem
    __shared__ float s_qx[BLOCK + 1], s_qy[BLOCK + 1]; // s_q*[i] = normalized quat of thread i's
    __shared__ float s_qz[BLOCK + 1], s_qw[BLOCK + 1]; //           first elem; [BLOCK] = halo
    __shared__ int   s_qv[BLOCK + 1];

    const int b   = blockIdx.y;          // 2D grid: no SALU div/mod
    const int t0  = blockIdx.x * CHUNK;
    const int tid = threadIdx.x;
    const int t   = t0 + tid * VEC;

    const float* imu_b = imu + (size_t)b * 7 * T;
    float*       out_b = out + (size_t)b * 16 * T;

    // ---- phase 1: wide loads, passthrough, gravity / linear_acc / lam ----
    v4f a0 = *(const v4f*)(imu_b + 0 * T + t);
    v4f a1 = *(const v4f*)(imu_b + 1 * T + t);
    v4f a2 = *(const v4f*)(imu_b + 2 * T + t);
    v4f qw = *(const v4f*)(imu_b + 3 * T + t);
    v4f qx = *(const v4f*)(imu_b + 4 * T + t);
    v4f qy = *(const v4f*)(imu_b + 5 * T + t);
    v4f qz = *(const v4f*)(imu_b + 6 * T + t);

    // channels 0..6: raw passthrough (non-temporal: never re-read)
    __builtin_nontemporal_store(a0, (v4f*)(out_b + 0 * T + t));
    __builtin_nontemporal_store(a1, (v4f*)(out_b + 1 * T + t));
    __builtin_nontemporal_store(a2, (v4f*)(out_b + 2 * T + t));
    __builtin_nontemporal_store(qw, (v4f*)(out_b + 3 * T + t));
    __builtin_nontemporal_store(qx, (v4f*)(out_b + 4 * T + t));
    __builtin_nontemporal_store(qy, (v4f*)(out_b + 5 * T + t));
    __builtin_nontemporal_store(qz, (v4f*)(out_b + 6 * T + t));

    float xn[VEC], yn[VEC], zn[VEC], wn[VEC], lam[VEC];
    bool  v1[VEC];
    v4f lax, lay, laz, lamv;
#pragma unroll
    for (int j = 0; j < VEC; ++j) {
        float x = qx[j], y = qy[j], z = qz[j], w = qw[j];
        float n   = sqrtf(x * x + y * y + z * z + w * w);
        bool  v   = n > TOLF;
        float inv = 1.0f / (v ? n : 1.0f);       // 1 div + 4 mul instead of 4 divs
        xn[j] = x * inv; yn[j] = y * inv; zn[j] = z * inv; wn[j] = w * inv;
        v1[j] = v;
        float gx = v ? 19.62f * (xn[j] * zn[j] - wn[j] * yn[j]) : 0.0f;
        float gy = v ? 19.62f * (wn[j] * xn[j] + yn[j] * zn[j]) : 0.0f;
        float gz = v ? (9.81f - 19.62f * (xn[j] * xn[j] + yn[j] * yn[j])) : 0.0f;
        lax[j] = a0[j] - gx;
        lay[j] = a1[j] - gy;
        laz[j] = a2[j] - gz;
        lam[j] = sqrtf(lax[j] * lax[j] + lay[j] * lay[j] + laz[j] * laz[j]);
        lamv[j] = lam[j];
    }
    __builtin_nontemporal_store(lax,  (v4f*)(out_b + 7  * T + t));
    __builtin_nontemporal_store(lay,  (v4f*)(out_b + 8  * T + t));
    __builtin_nontemporal_store(laz,  (v4f*)(out_b + 9  * T + t));
    __builtin_nontemporal_store(lamv, (v4f*)(out_b + 10 * T + t));

    // ---- halo exchange through LDS ----
    s_lam[tid + 1] = lam[VEC - 1];
    s_qx[tid] = xn[0]; s_qy[tid] = yn[0]; s_qz[tid] = zn[0]; s_qw[tid] = wn[0];
    s_qv[tid] = v1[0];

    if (tid == 0) {
        s_lam[0] = (t0 > 0) ? lam_at(imu_b, T, t0 - 1) : 0.0f;  // unused when t0==0
    }
    if (tid == BLOCK - 1) {
        int th = t0 + CHUNK;               // quat at one-past-chunk (shift_next clamps at T-1)
        if (th > T - 1) th = T - 1;
        float w = imu_b[3 * T + th], x = imu_b[4 * T + th];
        float y = imu_b[5 * T + th], z = imu_b[6 * T + th];
        float n   = sqrtf(x * x + y * y + z * z + w * w);
        bool  v   = n > TOLF;
        float inv = 1.0f / (v ? n : 1.0f);
        s_qx[BLOCK] = x * inv; s_qy[BLOCK] = y * inv;
        s_qz[BLOCK] = z * inv; s_qw[BLOCK] = w * inv;
        s_qv[BLOCK] = v;
    }
    __syncthreads();

    // ---- phase 2: jerk, angular velocity, angular distance ----
    const float lam_prev0 = s_lam[tid];
    const float hx = s_qx[tid + 1], hy = s_qy[tid + 1];
    const float hz = s_qz[tid + 1], hw = s_qw[tid + 1];
    const bool  hv = s_qv[tid + 1] != 0;

    const float INV_DT = 1.0f / DTF;     // compile-time constant fold

    v4f jerkv, wxv, wyv, wzv, adv;
#pragma unroll
    for (int j = 0; j < VEC; ++j) {
        float lp = (j == 0) ? lam_prev0 : lam[j - 1];
        jerkv[j] = (t + j == 0) ? 0.0f : (lam[j] - lp);

        float x2, y2, z2, w2; bool vb;
        if (j < VEC - 1) { x2 = xn[j + 1]; y2 = yn[j + 1]; z2 = zn[j + 1]; w2 = wn[j + 1]; vb = v1[j + 1]; }
        else             { x2 = hx;        y2 = hy;        z2 = hz;        w2 = hw;        vb = hv; }

        float dot = xn[j] * x2 + yn[j] * y2 + zn[j] * z2 + wn[j] * w2;
        float sgn = (dot < 0.0f) ? -1.0f : 1.0f;   // NaN -> +1, matches jnp.where
        x2 *= sgn; y2 *= sgn; z2 *= sgn; w2 *= sgn;

        float dx = wn[j] * x2 - xn[j] * w2 - yn[j] * z2 + zn[j] * y2;
        float dy = wn[j] * y2 + xn[j] * z2 - yn[j] * w2 - zn[j] * x2;
        float dz = wn[j] * z2 - xn[j] * y2 + yn[j] * x2 - zn[j] * w2;
        float dw = wn[j] * w2 + xn[j] * x2 + yn[j] * y2 + zn[j] * z2;

        float vn   = sqrtf(dx * dx + dy * dy + dz * dz);
        float angd = 2.0f * atan2f(vn, dw);        // unclipped (ang_dist)
        // dw = sgn*dot >= -eps, so clip(dw,-1,1) only binds at +1:
        // angle_v == angle_d unless dw > 1 (rare rounding case) -> divergent slow path.
        float angv = angd;
        if (dw > 1.0f) angv = 2.0f * atan2f(vn, 1.0f);

        bool  big     = vn > TOLF;
        float scaledt = big ? (angv / vn) * INV_DT : 0.0f;   // fold /DT into one factor
        float wx = dx * scaledt;
        float wy = dy * scaledt;
        float wz = dz * scaledt;
        bool vw = v1[j] && vb && finitef(wx) && finitef(wy) && finitef(wz);
        wxv[j] = vw ? wx : 0.0f;
        wyv[j] = vw ? wy : 0.0f;
        wzv[j] = vw ? wz : 0.0f;

        adv[j] = (v1[j] && vb && finitef(angd)) ? angd : 0.0f;
    }
    __builtin_nontemporal_store(jerkv, (v4f*)(out_b + 11 * T + t));
    __builtin_nontemporal_store(wxv,   (v4f*)(out_b + 12 * T + t));
    __builtin_nontemporal_store(wyv,   (v4f*)(out_b + 13 * T + t));
    __builtin_nontemporal_store(wzv,   (v4f*)(out_b + 14 * T + t));
    __builtin_nontemporal_store(adv,   (v4f*)(out_b + 15 * T + t));
}

extern "C" void kernel_launch(void* const* d_in, const int* in_sizes, int n_in,
                              void* d_out, int out_size, void* d_ws, size_t ws_size,
                              hipStream_t stream) {
    (void)n_in; (void)out_size; (void)d_ws; (void)ws_size;
    const float* imu = (const float*)d_in[0];
    float* out = (float*)d_out;

    const int T = 8192;                       // time axis (multiple of CHUNK=1024)
    const int B = in_sizes[0] / (7 * T);      // batch
    const int nchunk = T / CHUNK;

    dim3 grid(nchunk, B);                     // x = time chunk, y = batch (no div/mod in kernel)
    dim3 block(BLOCK);
    imu_feat_kernel<<<grid, block, 0, stream>>>(imu, out, T);
}